// KGNNLayer_44899588112534
// MI455X (gfx1250) — compile-verified
//
#include <hip/hip_runtime.h>
#include <hip/hip_bf16.h>

typedef __attribute__((ext_vector_type(2))) float v2f;
typedef __attribute__((ext_vector_type(4))) float v4f;
typedef __attribute__((ext_vector_type(8))) float v8f;
typedef unsigned int u32x4 __attribute__((ext_vector_type(4)));
typedef int          i32x4 __attribute__((ext_vector_type(4)));
typedef int          i32x8 __attribute__((ext_vector_type(8)));

#define TILE_M   16
#define DIM      128
#define KNBR     32
#define NREL     64
#define LSTR     132   // padded LDS row stride (floats): breaks 64-bank alias of 128

// ---------------------------------------------------------------------------
// TDM 2-D copy: global (rows x cols f32, row stride in elems) -> LDS, with
// hardware padding of 4 DWORDs after every 128 DWORDs (=> 132-float stride).
// D# layout per CDNA5 ISA 8.3/8.4. Groups 2/3 zero (<=2-D tensor).
// 6-arg builtin form: (g0, g1, g2, g3, g_extra, cpol).
// ---------------------------------------------------------------------------
__device__ __forceinline__ void tdm_load_2d_padded(const void* gptr, void* lptr,
                                                   unsigned rows, unsigned cols,
                                                   unsigned row_stride_elems) {
    const unsigned lds_off = (unsigned)(uintptr_t)lptr;           // LDS aperture: low 32 bits = offset
    const unsigned long long ga = (unsigned long long)(uintptr_t)gptr;

    u32x4 g0;
    g0.x = 1u;                                                    // count=1, is_restore=0, gather=0
    g0.y = lds_off;                                               // lds_addr
    g0.z = (unsigned)(ga & 0xFFFFFFFFu);                          // global_addr[31:0]
    g0.w = (unsigned)((ga >> 32) & 0x1FFFFFFu) | (2u << 30);      // global_addr[56:32], type=2

    i32x8 g1;
    // data_size=2 (4B), pad_enable=1, pad_interval=6 (128 DWORDs), pad_amount=3 (4 DWORDs)
    g1[0] = (int)((2u << 16) | (1u << 20) | (6u << 22) | (3u << 25));
    g1[1] = (int)((cols & 0xFFFFu) << 16);                        // tensor_dim0[15:0] @bits63:48
    g1[2] = (int)(((cols >> 16) & 0xFFFFu) | ((rows & 0xFFFFu) << 16)); // dim0 hi / dim1 lo
    g1[3] = (int)(((rows >> 16) & 0xFFFFu) | ((cols & 0xFFFFu) << 16)); // dim1 hi / tile_dim0
    g1[4] = (int)(rows & 0xFFFFu);                                // tile_dim1 ; tile_dim2=0
    g1[5] = (int)row_stride_elems;                                // tensor_dim0_stride[31:0]
    g1[6] = 0;                                                    // stride hi / dim1_stride lo
    g1[7] = 0;
    i32x4 z4 = {0, 0, 0, 0};
    i32x8 z8 = {0, 0, 0, 0, 0, 0, 0, 0};
    __builtin_amdgcn_tensor_load_to_lds(g0, g1, z4, z4, z8, 0);
}

__launch_bounds__(256, 2)
__global__ void kgnn_fused_kernel(const float* __restrict__ user_emb,
                                  const int*   __restrict__ entity_ids,
                                  const int*   __restrict__ neigh_ent_ids,
                                  const int*   __restrict__ neigh_rel_ids,
                                  const float* __restrict__ entity_table,
                                  const float* __restrict__ relation_table,
                                  const float* __restrict__ Wmat,
                                  float*       __restrict__ out)
{
    __shared__ float lds_user[TILE_M * LSTR];   // 16x128 user tile
    __shared__ float lds_W   [DIM    * LSTR];   // full 128x128 W
    __shared__ float lds_rel [NREL   * LSTR];   // full 64x128 relation table
    __shared__ float lds_proj[TILE_M * LSTR];   // 16x128 user_proj tile

    const int tid  = threadIdx.x;
    const int lane = tid & 31;
    const int wave = tid >> 5;            // 0..7
    const int row0 = blockIdx.x * TILE_M; // first global row of this tile

    // ---- hoisted index loads (independent of LDS) + gather prefetch ------
    int rid[2], nid[2], sid[2];
    #pragma unroll
    for (int rr = 0; rr < 2; ++rr) {
        const size_t b = (size_t)row0 + wave + 8 * rr;
        rid[rr] = neigh_rel_ids[b * KNBR + lane];
        nid[rr] = neigh_ent_ids[b * KNBR + lane];
        sid[rr] = entity_ids[b];
    }
    #pragma unroll
    for (int rr = 0; rr < 2; ++rr) {
        const float* nrow = entity_table + (size_t)nid[rr] * DIM;
        __builtin_prefetch(nrow,       0, 3);   // 4 x 128B lines cover the 512B row
        __builtin_prefetch(nrow + 32,  0, 3);
        __builtin_prefetch(nrow + 64,  0, 3);
        __builtin_prefetch(nrow + 96,  0, 3);
        __builtin_prefetch(entity_table + (size_t)sid[rr] * DIM + 4 * lane, 0, 3);
    }

    // ---- TDM staging: one wave issues 3 DMA descriptors ------------------
    if (wave == 0) {
        tdm_load_2d_padded(user_emb + (size_t)row0 * DIM, lds_user, TILE_M, DIM, DIM);
        tdm_load_2d_padded(Wmat,            lds_W,   DIM,  DIM, DIM);
        tdm_load_2d_padded(relation_table,  lds_rel, NREL, DIM, DIM);
        __builtin_amdgcn_s_wait_tensorcnt(0);
    }
    __syncthreads();

    // ---------------- Phase 1: user_proj = user_emb @ W^T (WMMA) ----------
    // A (16x4 f32): lane%16 = M, K = 2*(lane/16) + j   (j = vgpr 0/1)
    // B (4x16 f32): lane%16 = N, K = 2*(lane/16) + j ; B[k][n] = W[n][k]
    {
        const int m = lane & 15;
        const int h = lane >> 4;
        const float* aBase = &lds_user[m * LSTR + 2 * h];
        const float* bBase = &lds_W[(wave * 16 + m) * LSTR + 2 * h];
        v8f acc = {};
        #pragma unroll
        for (int t = 0; t < DIM / 4; ++t) {            // 32 WMMA ops
            v2f a = *(const v2f*)(aBase + 4 * t);
            v2f b = *(const v2f*)(bBase + 4 * t);
            acc = __builtin_amdgcn_wmma_f32_16x16x4_f32(
                      false, a, false, b, (short)0, acc, false, false);
        }
        // D layout: vgpr r, lane L -> (M = r + 8*(L/16), N = L%16)
        #pragma unroll
        for (int r = 0; r < 8; ++r) {
            int M = r + 8 * h;
            lds_proj[M * LSTR + wave * 16 + m] = acc[r];
        }
    }
    __syncthreads();

    // ---------------- Phases 2-4: per-row (wave handles 2 rows) -----------
    #pragma unroll
    for (int rr = 0; rr < 2; ++rr) {
        const int   lrow = wave + 8 * rr;        // local row 0..15
        const size_t b   = (size_t)row0 + lrow;  // global row

        // lane = neighbor index k: score = <user_proj[b], relation_table[rid]>
        float s = 0.f;
        const float* up = &lds_proj[lrow   * LSTR];
        const float* rp = &lds_rel [rid[rr] * LSTR];
        #pragma unroll 8
        for (int d4 = 0; d4 < DIM / 4; ++d4) {
            v4f u = *(const v4f*)(up + 4 * d4);   // broadcast (same addr all lanes)
            v4f r = *(const v4f*)(rp + 4 * d4);
            s += u.x * r.x + u.y * r.y + u.z * r.z + u.w * r.w;
        }

        // softmax over the 32 lanes (K = 32)
        float mx = s;
        #pragma unroll
        for (int off = 16; off > 0; off >>= 1)
            mx = fmaxf(mx, __shfl_xor(mx, off, 32));
        float e = __expf(s - mx);
        float sum = e;
        #pragma unroll
        for (int off = 16; off > 0; off >>= 1)
            sum += __shfl_xor(sum, off, 32);
        const float wgt = e / sum;

        // lane role switch: lane covers dims [4*lane, 4*lane+4)
        v4f acc = {0.f, 0.f, 0.f, 0.f};
        #pragma unroll 8
        for (int k = 0; k < KNBR; ++k) {
            const float wk = __shfl(wgt,     k, 32);
            const int   nk = __shfl(nid[rr], k, 32);
            v4f v = *(const v4f*)(entity_table + (size_t)nk * DIM + 4 * lane);
            acc.x += wk * v.x;
            acc.y += wk * v.y;
            acc.z += wk * v.z;
            acc.w += wk * v.w;
        }

        v4f sv = *(const v4f*)(entity_table + (size_t)sid[rr] * DIM + 4 * lane);
        v4f o;
        o.x = fmaxf(sv.x + acc.x, 0.f);
        o.y = fmaxf(sv.y + acc.y, 0.f);
        o.z = fmaxf(sv.z + acc.z, 0.f);
        o.w = fmaxf(sv.w + acc.w, 0.f);
        *(v4f*)(out + b * DIM + 4 * lane) = o;
    }
}

extern "C" void kernel_launch(void* const* d_in, const int* in_sizes, int n_in,
                              void* d_out, int out_size, void* d_ws, size_t ws_size,
                              hipStream_t stream) {
    const float* user_emb       = (const float*)d_in[0];
    const int*   entity_ids     = (const int*)  d_in[1];
    const int*   neigh_ent_ids  = (const int*)  d_in[2];
    const int*   neigh_rel_ids  = (const int*)  d_in[3];
    const float* entity_table   = (const float*)d_in[4];
    const float* relation_table = (const float*)d_in[5];
    const float* Wmat           = (const float*)d_in[6];
    float*       out            = (float*)d_out;

    const int B = in_sizes[0] / DIM;          // 16384
    const int grid = B / TILE_M;              // 1024 blocks, 16 rows each

    kgnn_fused_kernel<<<grid, 256, 0, stream>>>(
        user_emb, entity_ids, neigh_ent_ids, neigh_rel_ids,
        entity_table, relation_table, Wmat, out);
}